// CustomSoftmax_81604378624394
// MI455X (gfx1250) — compile-verified
//
#include <hip/hip_runtime.h>
#include <math.h>
#include <stdint.h>

// Row-wise masked softmax, B=8192 rows x L=4096 fp32.
// Memory-bound: 256 MB min traffic -> ~11us at 23.3 TB/s.
// One block per row: async-copy the row to LDS (gfx1250 async engine),
// then 3 passes out of registers, NT stores for the single-use output.

typedef __attribute__((ext_vector_type(4))) float v4f;

#define TILE_L   4096
#define THREADS  256
#define NWAVES   (THREADS / 32)
#define CHUNKS   4   // 4 chunks * 256 threads * 16B = 16 KB = 4096 floats

__global__ __launch_bounds__(THREADS)
void masked_softmax_rows(const float* __restrict__ X,
                         const int*   __restrict__ Nlen,
                         float*       __restrict__ Out)
{
    __shared__ __align__(16) float smem[TILE_L];
    __shared__ float red_max[NWAVES];
    __shared__ float red_sum[NWAVES];

    const int row = blockIdx.x;
    const int tid = threadIdx.x;
    const int n   = Nlen[row];                 // valid length, uniform per block

    const float* Xrow = X   + (size_t)row * TILE_L;
    float*       Orow = Out + (size_t)row * TILE_L;

    // ---- Stage the row into LDS via the gfx1250 async copy engine ----
    // Low 32 bits of a flat shared pointer are the LDS byte offset
    // (flat->LDS aperture truncation), which is what VDST expects.
    unsigned           lds_base = (unsigned)(uintptr_t)smem;
    unsigned long long gbase    = (unsigned long long)(uintptr_t)Xrow;
#pragma unroll
    for (int c = 0; c < CHUNKS; ++c) {
        unsigned           off  = (unsigned)((c * THREADS + tid) * 16);
        unsigned           ldsa = lds_base + off;
        unsigned long long ga   = gbase + off;
        asm volatile("global_load_async_to_lds_b128 %0, %1, off"
                     :: "v"(ldsa), "v"(ga) : "memory");
    }
    asm volatile("s_wait_asynccnt 0" ::: "memory");
    __syncthreads();

    // ---- Pull this thread's 16 elements into registers (ds_load_b128 x4) ----
    v4f x[CHUNKS];
    const v4f* sm4 = (const v4f*)smem;
#pragma unroll
    for (int c = 0; c < CHUNKS; ++c)
        x[c] = sm4[c * THREADS + tid];

    // ---- Pass 1: masked row max ----
    float m = -INFINITY;
#pragma unroll
    for (int c = 0; c < CHUNKS; ++c) {
        int j0 = (c * THREADS + tid) * 4;
        m = fmaxf(m, (j0 + 0 < n) ? x[c].x : -INFINITY);
        m = fmaxf(m, (j0 + 1 < n) ? x[c].y : -INFINITY);
        m = fmaxf(m, (j0 + 2 < n) ? x[c].z : -INFINITY);
        m = fmaxf(m, (j0 + 3 < n) ? x[c].w : -INFINITY);
    }
#pragma unroll
    for (int off = 16; off > 0; off >>= 1)
        m = fmaxf(m, __shfl_xor(m, off, 32));

    const int wave = tid >> 5;
    const int lane = tid & 31;
    if (lane == 0) red_max[wave] = m;
    __syncthreads();
    float rowmax = red_max[0];
#pragma unroll
    for (int w = 1; w < NWAVES; ++w) rowmax = fmaxf(rowmax, red_max[w]);

    // ---- Pass 2: exp(x - max), masked sum ----
    v4f e[CHUNKS];
    float s = 0.0f;
#pragma unroll
    for (int c = 0; c < CHUNKS; ++c) {
        int j0 = (c * THREADS + tid) * 4;
        e[c].x = __expf(x[c].x - rowmax);
        e[c].y = __expf(x[c].y - rowmax);
        e[c].z = __expf(x[c].z - rowmax);
        e[c].w = __expf(x[c].w - rowmax);
        if (j0 + 0 < n) s += e[c].x;
        if (j0 + 1 < n) s += e[c].y;
        if (j0 + 2 < n) s += e[c].z;
        if (j0 + 3 < n) s += e[c].w;
    }
#pragma unroll
    for (int off = 16; off > 0; off >>= 1)
        s += __shfl_xor(s, off, 32);
    if (lane == 0) red_sum[wave] = s;
    __syncthreads();
    float rowsum = 0.0f;
#pragma unroll
    for (int w = 0; w < NWAVES; ++w) rowsum += red_sum[w];

    const float inv = 1.0f / rowsum;   // one accurate divide per thread

    // ---- Pass 3: masked scale + non-temporal store (output is single-use) ----
#pragma unroll
    for (int c = 0; c < CHUNKS; ++c) {
        int j0 = (c * THREADS + tid) * 4;
        v4f o;
        o.x = (j0 + 0 < n) ? e[c].x * inv : 0.0f;
        o.y = (j0 + 1 < n) ? e[c].y * inv : 0.0f;
        o.z = (j0 + 2 < n) ? e[c].z * inv : 0.0f;
        o.w = (j0 + 3 < n) ? e[c].w * inv : 0.0f;
        __builtin_nontemporal_store(o, (v4f*)(Orow + j0));
    }
}

extern "C" void kernel_launch(void* const* d_in, const int* in_sizes, int n_in,
                              void* d_out, int out_size, void* d_ws, size_t ws_size,
                              hipStream_t stream)
{
    const float* X    = (const float*)d_in[0];   // [B, L] fp32
    const int*   Nlen = (const int*)d_in[1];     // [B, 1] int32
    float*       Out  = (float*)d_out;           // [B, L] fp32

    const int rows = in_sizes[1];                // B (N has one entry per row)
    masked_softmax_rows<<<rows, THREADS, 0, stream>>>(X, Nlen, Out);
}